// GAT_FIXED_W_73504070304133
// MI455X (gfx1250) — compile-verified
//
#include <hip/hip_runtime.h>
#include <hip/hip_bf16.h>

#define N_NODES 50000
#define N_EDGES 800000
#define DN 64
#define DE 16
#define DO 64

typedef __attribute__((ext_vector_type(2))) float v2f;
typedef __attribute__((ext_vector_type(8))) float v8f;

// ---------------------------------------------------------------------------
// K1: per-node scalar projections + init. One wave32 per node.
//   s_src[n] = dot(n_feats[n], a_w[0:64])
//   s_dst[n] = dot(n_feats[n], a_w[80:144])
//   m[n] = -inf, denom[n] = 0, out row zeroed.
// ---------------------------------------------------------------------------
__global__ void node_init_kernel(const float* __restrict__ nf,
                                 const float* __restrict__ aw,
                                 float* __restrict__ ssrc,
                                 float* __restrict__ sdst,
                                 float* __restrict__ m,
                                 float* __restrict__ denom,
                                 float* __restrict__ out,
                                 int n_nodes) {
    const int wavesPerBlock = blockDim.x >> 5;
    const int node = blockIdx.x * wavesPerBlock + (threadIdx.x >> 5);
    const int lane = threadIdx.x & 31;
    if (node >= n_nodes) return;

    const float x0 = nf[node * DN + lane];
    const float x1 = nf[node * DN + 32 + lane];
    float ps = x0 * aw[lane]            + x1 * aw[32 + lane];
    float pd = x0 * aw[DN + DE + lane]  + x1 * aw[DN + DE + 32 + lane];
#pragma unroll
    for (int off = 16; off > 0; off >>= 1) {
        ps += __shfl_xor(ps, off, 32);
        pd += __shfl_xor(pd, off, 32);
    }
    out[node * DO + lane]      = 0.0f;
    out[node * DO + 32 + lane] = 0.0f;
    if (lane == 0) {
        ssrc[node]  = ps;
        sdst[node]  = pd;
        m[node]     = -__builtin_inff();
        denom[node] = 0.0f;
    }
}

// ---------------------------------------------------------------------------
// K2: ft = n_feats @ W via V_WMMA_F32_16X16X4_F32.
// One wave per 16x16 output tile; K=64 in steps of 4.
// A 16x4 layout (ISA 7.12.2): lanes 0-15 -> K=2*half..2*half+1 rows M=lane&15.
// B 4x16 layout: VGPR j, lanes 0-15 row K=2*half+j, col N=lane&15.
// C/D 16x16: VGPR j holds row (j + 8*half), col lane&15.
// ---------------------------------------------------------------------------
__global__ void gemm_ft_kernel(const float* __restrict__ A,   // [M x 64]
                               const float* __restrict__ W,   // [64 x 64]
                               float* __restrict__ ft,        // [M x 64]
                               int M) {
    const int wavesPerBlock = blockDim.x >> 5;
    const int wave = blockIdx.x * wavesPerBlock + (threadIdx.x >> 5);
    const int lane = threadIdx.x & 31;
    const int colTiles = DO / 16;            // 4
    const int rowTile = wave / colTiles;
    const int colTile = wave - rowTile * colTiles;
    if (rowTile * 16 >= M) return;           // wave-uniform: EXEC stays full

    const int m0   = rowTile * 16;
    const int n0   = colTile * 16;
    const int half = lane >> 4;              // 0 or 1
    const int lr   = lane & 15;

    v8f c = {};
#pragma unroll
    for (int k = 0; k < DN; k += 4) {
        const int ka = k + 2 * half;
        v2f a, b;
        a.x = A[(m0 + lr) * DN + ka];
        a.y = A[(m0 + lr) * DN + ka + 1];
        b.x = W[ka * DO + n0 + lr];
        b.y = W[(ka + 1) * DO + n0 + lr];
        c = __builtin_amdgcn_wmma_f32_16x16x4_f32(
                /*neg_a=*/false, a, /*neg_b=*/false, b,
                /*c_mod=*/(short)0, c, /*reuse_a=*/false, /*reuse_b=*/false);
    }
#pragma unroll
    for (int j = 0; j < 8; ++j) {
        ft[(m0 + j + 8 * half) * DO + n0 + lr] = c[j];
    }
}

// ---------------------------------------------------------------------------
// K3: edge scores + float segment-max via native int atomics.
// ---------------------------------------------------------------------------
__global__ void edge_scores_kernel(const float* __restrict__ ef,
                                   const float* __restrict__ aw,
                                   const float* __restrict__ ssrc,
                                   const float* __restrict__ sdst,
                                   const int* __restrict__ src,
                                   const int* __restrict__ dst,
                                   float* __restrict__ scores,
                                   float* __restrict__ m,
                                   int n_edges) {
    const int e = blockIdx.x * blockDim.x + threadIdx.x;
    if (e >= n_edges) return;

    const float4* row = (const float4*)(ef + (size_t)e * DE);
    float acc = 0.0f;
#pragma unroll
    for (int i = 0; i < DE / 4; ++i) {
        const float4 v = row[i];
        acc += v.x * aw[DN + i * 4 + 0] + v.y * aw[DN + i * 4 + 1]
             + v.z * aw[DN + i * 4 + 2] + v.w * aw[DN + i * 4 + 3];
    }
    const int d = dst[e];
    const float s = ssrc[src[e]] + acc + sdst[d];
    scores[e] = s;

    // float max via sign-split int atomics (monotone bit encodings)
    float* addr = m + d;
    if (s >= 0.0f) atomicMax((int*)addr, __float_as_int(s));
    else           atomicMin((unsigned int*)addr, __float_as_uint(s));
}

// ---------------------------------------------------------------------------
// K4: m = isfinite(m) ? m : 0
// ---------------------------------------------------------------------------
__global__ void fin_m_kernel(float* __restrict__ m, int n) {
    const int i = blockIdx.x * blockDim.x + threadIdx.x;
    if (i >= n) return;
    const float v = m[i];
    if (!__builtin_isfinite(v)) m[i] = 0.0f;
}

// ---------------------------------------------------------------------------
// K5: ex = exp(score - m[dst]) (in place); denom[dst] += ex
// ---------------------------------------------------------------------------
__global__ void edge_exp_kernel(float* __restrict__ scores_ex,
                                const float* __restrict__ m,
                                const int* __restrict__ dst,
                                float* __restrict__ denom,
                                int n_edges) {
    const int e = blockIdx.x * blockDim.x + threadIdx.x;
    if (e >= n_edges) return;
    const int d = dst[e];
    const float v = __expf(scores_ex[e] - m[d]);
    scores_ex[e] = v;
    atomicAdd(denom + d, v);
}

// ---------------------------------------------------------------------------
// K6: out[dst] += ft[src] * alpha  — one thread per (edge, feature).
// Output (12.8 MB) is L2-resident; f32 adds are native VMEM atomics.
// ---------------------------------------------------------------------------
__global__ void edge_accum_kernel(const float* __restrict__ ft,
                                  const float* __restrict__ ex,
                                  const float* __restrict__ denom,
                                  const int* __restrict__ src,
                                  const int* __restrict__ dst,
                                  float* __restrict__ out,
                                  int n_edges) {
    const int tid = blockIdx.x * blockDim.x + threadIdx.x;
    if (tid >= n_edges * DO) return;
    const int e = tid >> 6;       // DO == 64
    const int j = tid & (DO - 1);
    const int d = dst[e];
    const float alpha = ex[e] / fmaxf(denom[d], 1e-9f);
    atomicAdd(out + (size_t)d * DO + j, ft[(size_t)src[e] * DO + j] * alpha);
}

// ---------------------------------------------------------------------------
// K7: leaky ReLU in place.
// ---------------------------------------------------------------------------
__global__ void lrelu_kernel(float* __restrict__ out, int n) {
    const int i = blockIdx.x * blockDim.x + threadIdx.x;
    if (i >= n) return;
    const float v = out[i];
    out[i] = (v > 0.0f) ? v : 0.01f * v;
}

// ---------------------------------------------------------------------------
extern "C" void kernel_launch(void* const* d_in, const int* in_sizes, int n_in,
                              void* d_out, int out_size, void* d_ws, size_t ws_size,
                              hipStream_t stream) {
    const float* n_feats = (const float*)d_in[0];   // [50000 x 64]
    const float* e_feats = (const float*)d_in[1];   // [800000 x 16]
    const float* W       = (const float*)d_in[2];   // [64 x 64]
    const float* a_w     = (const float*)d_in[3];   // [144]
    const int*   src     = (const int*)d_in[4];     // [800000]
    const int*   dst     = (const int*)d_in[5];     // [800000]
    float*       out     = (float*)d_out;           // [50000 x 64]

    // workspace layout (floats)
    float* ws     = (float*)d_ws;
    float* ft     = ws;                                   // 3,200,000
    float* ssrc   = ft + (size_t)N_NODES * DO;            //    50,000
    float* sdst   = ssrc + N_NODES;                       //    50,000
    float* m      = sdst + N_NODES;                       //    50,000
    float* denom  = m + N_NODES;                          //    50,000
    float* scores = denom + N_NODES;                      //   800,000
    (void)ws_size; (void)in_sizes; (void)n_in; (void)out_size;

    const int THREADS = 256;
    const int wavesPerBlock = THREADS / 32;

    // K1: node scalars + init (wave per node)
    {
        int blocks = (N_NODES + wavesPerBlock - 1) / wavesPerBlock;
        node_init_kernel<<<blocks, THREADS, 0, stream>>>(
            n_feats, a_w, ssrc, sdst, m, denom, out, N_NODES);
    }
    // K2: WMMA GEMM ft = n_feats @ W
    {
        int tiles = (N_NODES / 16) * (DO / 16);           // 3125 * 4 = 12500 waves
        int blocks = (tiles + wavesPerBlock - 1) / wavesPerBlock;
        gemm_ft_kernel<<<blocks, THREADS, 0, stream>>>(n_feats, W, ft, N_NODES);
    }
    // K3: edge scores + segment max
    {
        int blocks = (N_EDGES + THREADS - 1) / THREADS;
        edge_scores_kernel<<<blocks, THREADS, 0, stream>>>(
            e_feats, a_w, ssrc, sdst, src, dst, scores, m, N_EDGES);
    }
    // K4: finalize m
    {
        int blocks = (N_NODES + THREADS - 1) / THREADS;
        fin_m_kernel<<<blocks, THREADS, 0, stream>>>(m, N_NODES);
    }
    // K5: exp + denom
    {
        int blocks = (N_EDGES + THREADS - 1) / THREADS;
        edge_exp_kernel<<<blocks, THREADS, 0, stream>>>(scores, m, dst, denom, N_EDGES);
    }
    // K6: weighted scatter-add
    {
        int total = N_EDGES * DO;                         // 51.2M
        int blocks = (total + THREADS - 1) / THREADS;
        edge_accum_kernel<<<blocks, THREADS, 0, stream>>>(
            ft, scores, denom, src, dst, out, N_EDGES);
    }
    // K7: leaky ReLU
    {
        int total = N_NODES * DO;
        int blocks = (total + THREADS - 1) / THREADS;
        lrelu_kernel<<<blocks, THREADS, 0, stream>>>(out, total);
    }
}